// Decoder_49314814493291
// MI455X (gfx1250) — compile-verified
//
#include <hip/hip_runtime.h>
#include <math.h>

// Problem constants (match reference)
#define VSZ 10000
#define ESZ 512
#define HSZ 1024
#define LSZ 2
#define TSZ 48
#define SSZ 48
#define BSZ 32
#define PAD_ID 0
#define COPY_ID 1
#define EPS_C 1e-07f

typedef __attribute__((ext_vector_type(2))) float v2f;
typedef __attribute__((ext_vector_type(8))) float v8f;

// ---------------------------------------------------------------------------
// WMMA f32 2Mx2N register-blocked helper:
//   c{m}{n} += A{m}_tile(16xK) * B{n}_tile(16xK)^T
// 4 b64 loads feed 4 WMMAs per K=4 step (both operands reused twice).
// Layout per CDNA5 ISA 7.12.2 (32-bit A 16x4 / B 4x16 / 32-bit C/D 16x16).
// ---------------------------------------------------------------------------
__device__ __forceinline__ void wmma_tile4_f32(const float* __restrict__ A0,
                                               const float* __restrict__ A1,
                                               int lda,
                                               const float* __restrict__ B0,
                                               const float* __restrict__ B1,
                                               int ldb, int K,
                                               v8f& c00, v8f& c01,
                                               v8f& c10, v8f& c11) {
    const int lane = threadIdx.x & 31;
    const int r    = lane & 15;           // m for A, n for B
    const int ksel = (lane >> 4) << 1;    // lanes 16-31 hold K+2,K+3
    const float* a0p = A0 + (long)r * lda + ksel;
    const float* a1p = A1 + (long)r * lda + ksel;
    const float* b0p = B0 + (long)r * ldb + ksel;
    const float* b1p = B1 + (long)r * ldb + ksel;
#pragma unroll 4
    for (int k = 0; k < K; k += 4) {
        v2f b0 = *(const v2f*)(b0p + k);
        v2f b1 = *(const v2f*)(b1p + k);
        v2f a0 = *(const v2f*)(a0p + k);
        v2f a1 = *(const v2f*)(a1p + k);
        c00 = __builtin_amdgcn_wmma_f32_16x16x4_f32(false, a0, false, b0,
                                                    (short)0, c00, false, false);
        c10 = __builtin_amdgcn_wmma_f32_16x16x4_f32(false, a1, false, b0,
                                                    (short)0, c10, false, false);
        c01 = __builtin_amdgcn_wmma_f32_16x16x4_f32(false, a0, false, b1,
                                                    (short)0, c01, false, false);
        c11 = __builtin_amdgcn_wmma_f32_16x16x4_f32(false, a1, false, b1,
                                                    (short)0, c11, false, false);
    }
}

// Store one 32x16 (two M tiles) column block with bias.
__device__ __forceinline__ void store_tile2(float* __restrict__ C, int ldc,
                                            int m0, int n0, float bv,
                                            const v8f& acc0, const v8f& acc1) {
    const int lane = threadIdx.x & 31;
    const int n  = n0 + (lane & 15);
    const int mb = m0 + ((lane >> 4) << 3);
#pragma unroll
    for (int r = 0; r < 8; ++r)
        C[(long)(mb + r) * ldc + n] = acc0[r] + bv;
#pragma unroll
    for (int r = 0; r < 8; ++r)
        C[(long)(mb + 16 + r) * ldc + n] = acc1[r] + bv;
}

// C[m0:32, n0:32] = A*B^T + bias  (one wave per 32x32 tile; blockDim.x == 32)
// Handles odd N-tile counts (e.g. V=10000) by clamping the 2nd N tile.
__global__ void gemm_bias_kernel(const float* __restrict__ A, int lda,
                                 const float* __restrict__ Bm, int ldb,
                                 const float* __restrict__ bias,
                                 float* __restrict__ C, int ldc, int K, int N) {
    const int n0 = blockIdx.x * 32;
    const int m0 = blockIdx.y * 32;
    const bool dup = (n0 + 16 >= N);          // wave-uniform tail clamp
    const int n1 = dup ? n0 : n0 + 16;
    v8f c00 = {}, c01 = {}, c10 = {}, c11 = {};
    wmma_tile4_f32(A + (long)m0 * lda, A + (long)(m0 + 16) * lda, lda,
                   Bm + (long)n0 * ldb, Bm + (long)n1 * ldb, ldb, K,
                   c00, c01, c10, c11);
    const int l15 = (threadIdx.x & 31) & 15;
    const float bv0 = bias ? bias[n0 + l15] : 0.0f;
    store_tile2(C, ldc, m0, n0, bv0, c00, c10);
    if (!dup) {
        const float bv1 = bias ? bias[n1 + l15] : 0.0f;
        store_tile2(C, ldc, m0, n1, bv1, c01, c11);
    }
}

// Fused LSTM gate pre-activation: G = X*Wi^T + Hs*Wh^T + bi + bh  (M = B = 32)
__global__ void gates_kernel(const float* __restrict__ X, int ldx,
                             const float* __restrict__ Wi, int Ki,
                             const float* __restrict__ Hs,
                             const float* __restrict__ Wh,
                             const float* __restrict__ bi,
                             const float* __restrict__ bh,
                             float* __restrict__ G) {
    const int n0 = blockIdx.x * 32;
    const int n1 = n0 + 16;                   // N = 4H, always even tile count
    v8f c00 = {}, c01 = {}, c10 = {}, c11 = {};
    wmma_tile4_f32(X, X + (long)16 * ldx, ldx,
                   Wi + (long)n0 * Ki, Wi + (long)n1 * Ki, Ki, Ki,
                   c00, c01, c10, c11);
    wmma_tile4_f32(Hs, Hs + (long)16 * HSZ, HSZ,
                   Wh + (long)n0 * HSZ, Wh + (long)n1 * HSZ, HSZ, HSZ,
                   c00, c01, c10, c11);
    const int l15 = (threadIdx.x & 31) & 15;
    store_tile2(G, 4 * HSZ, 0, n0, bi[n0 + l15] + bh[n0 + l15], c00, c10);
    store_tile2(G, 4 * HSZ, 0, n1, bi[n1 + l15] + bh[n1 + l15], c01, c11);
}

// ---------------------------------------------------------------------------
// Elementwise / attention kernels
// ---------------------------------------------------------------------------
__global__ void init_state_kernel(const float* __restrict__ h0,
                                  const float* __restrict__ c0,
                                  float* __restrict__ hs, float* __restrict__ cs,
                                  float* __restrict__ feed) {
    int i = blockIdx.x * blockDim.x + threadIdx.x;
    if (i < LSZ * BSZ * HSZ) { hs[i] = h0[i]; cs[i] = c0[i]; }
    if (i < BSZ * HSZ) feed[i] = 0.0f;
}

__global__ void embed_concat_kernel(const int* __restrict__ ref_tokens, int t,
                                    const float* __restrict__ embed,
                                    const float* __restrict__ feed,
                                    float* __restrict__ X) {
    const int b = blockIdx.x;
    const int tok = ref_tokens[t * BSZ + b];
    float* xr = X + (long)b * (ESZ + HSZ);
    for (int e = threadIdx.x; e < ESZ; e += blockDim.x)
        xr[e] = embed[(long)tok * ESZ + e];
    for (int j = threadIdx.x; j < HSZ; j += blockDim.x)
        xr[ESZ + j] = feed[b * HSZ + j];
}

__device__ __forceinline__ float sigf(float x) { return 1.0f / (1.0f + expf(-x)); }

__global__ void lstm_update_kernel(const float* __restrict__ G,
                                   float* __restrict__ h, float* __restrict__ c) {
    int idx = blockIdx.x * blockDim.x + threadIdx.x;   // B*H threads
    int b = idx / HSZ, j = idx % HSZ;
    const float* g = G + (long)b * 4 * HSZ;
    float ig = sigf(g[j]);
    float fg = sigf(g[HSZ + j]);
    float gg = tanhf(g[2 * HSZ + j]);
    float og = sigf(g[3 * HSZ + j]);
    float cn = fg * c[idx] + ig * gg;
    c[idx] = cn;
    h[idx] = og * tanhf(cn);
}

// scores[s,b] = attk[s,b,:] . hidden[b,:] + mask  (one wave per (s,b))
__global__ void attn_scores_kernel(const float* __restrict__ attk,
                                   const float* __restrict__ hidden,
                                   const int* __restrict__ src_tokens,
                                   float* __restrict__ scores) {
    int wid  = (blockIdx.x * blockDim.x + threadIdx.x) >> 5;
    int lane = threadIdx.x & 31;
    if (wid >= SSZ * BSZ) return;
    int s = wid / BSZ, b = wid % BSZ;
    const float* kp = attk + (long)(s * BSZ + b) * HSZ;
    const float* hp = hidden + (long)b * HSZ;
    float sum = 0.0f;
    for (int k = lane; k < HSZ; k += 32) sum += kp[k] * hp[k];
#pragma unroll
    for (int off = 16; off > 0; off >>= 1) sum += __shfl_xor(sum, off, 32);
    if (lane == 0) {
        float m = (src_tokens[s * BSZ + b] == PAD_ID) ? -99999.0f : 0.0f;
        scores[s * BSZ + b] = sum + m;
    }
}

// softmax over S, summary = dist . enc, cat = [hidden | summary] (block per b)
__global__ void attn_combine_kernel(float* __restrict__ dist,
                                    const float* __restrict__ enc,
                                    const float* __restrict__ hidden,
                                    float* __restrict__ cat) {
    const int b = blockIdx.x;
    __shared__ float sd[SSZ];
    if (threadIdx.x < SSZ) sd[threadIdx.x] = dist[threadIdx.x * BSZ + b];
    __syncthreads();
    if (threadIdx.x == 0) {
        float mx = sd[0];
        for (int s = 1; s < SSZ; ++s) mx = fmaxf(mx, sd[s]);
        float sum = 0.0f;
        for (int s = 0; s < SSZ; ++s) { sd[s] = expf(sd[s] - mx); sum += sd[s]; }
        float inv = 1.0f / sum;
        for (int s = 0; s < SSZ; ++s) sd[s] *= inv;
    }
    __syncthreads();
    if (threadIdx.x < SSZ) dist[threadIdx.x * BSZ + b] = sd[threadIdx.x];
    for (int j = threadIdx.x; j < HSZ; j += blockDim.x) {
        float acc = 0.0f;
        for (int s = 0; s < SSZ; ++s)
            acc += sd[s] * enc[(long)(s * BSZ + b) * HSZ + j];
        cat[(long)b * 2 * HSZ + HSZ + j] = acc;
        cat[(long)b * 2 * HSZ + j]       = hidden[b * HSZ + j];
    }
}

// copy_probs[b,:] = EPS; then serial scatter of dist (deterministic, no atomics)
__global__ void copy_probs_kernel(const int* __restrict__ src_tokens,
                                  const float* __restrict__ dist,
                                  float* __restrict__ copyp) {
    const int b = blockIdx.x;
    float* cp = copyp + (long)b * VSZ;
    for (int v = threadIdx.x; v < VSZ; v += blockDim.x) cp[v] = EPS_C;
    __syncthreads();
    if (threadIdx.x == 0)
        for (int s = 0; s < SSZ; ++s)
            cp[src_tokens[s * BSZ + b]] += dist[s * BSZ + b];
}

// softmax over V + copy blend + log (block per b)
__global__ void final_out_kernel(const float* __restrict__ plog,
                                 const float* __restrict__ copyp,
                                 float* __restrict__ out) {
    const int b = blockIdx.x, tid = threadIdx.x;
    __shared__ float red[256];
    const float* lp = plog + (long)b * VSZ;
    float mx = -3.4e38f;
    for (int v = tid; v < VSZ; v += 256) mx = fmaxf(mx, lp[v]);
    red[tid] = mx; __syncthreads();
    for (int s = 128; s > 0; s >>= 1) {
        if (tid < s) red[tid] = fmaxf(red[tid], red[tid + s]);
        __syncthreads();
    }
    const float m = red[0]; __syncthreads();
    float sm = 0.0f;
    for (int v = tid; v < VSZ; v += 256) sm += expf(lp[v] - m);
    red[tid] = sm; __syncthreads();
    for (int s = 128; s > 0; s >>= 1) {
        if (tid < s) red[tid] += red[tid + s];
        __syncthreads();
    }
    const float inv = 1.0f / red[0];
    const float cw  = expf(lp[COPY_ID] - m) * inv;
    const float omc = 1.0f - cw;
    for (int v = tid; v < VSZ; v += 256) {
        float p = expf(lp[v] - m) * inv;
        out[(long)b * VSZ + v] = logf(cw * copyp[(long)b * VSZ + v] + omc * p);
    }
}

// ---------------------------------------------------------------------------
extern "C" void kernel_launch(void* const* d_in, const int* in_sizes, int n_in,
                              void* d_out, int out_size, void* d_ws, size_t ws_size,
                              hipStream_t stream) {
    (void)in_sizes; (void)n_in; (void)out_size; (void)ws_size;
    const int*   ref_tokens = (const int*)  d_in[0];
    const int*   src_tokens = (const int*)  d_in[1];
    const float* enc        = (const float*)d_in[2];
    const float* h0         = (const float*)d_in[3];
    const float* c0         = (const float*)d_in[4];
    const float* embed      = (const float*)d_in[5];
    const float* Wk         = (const float*)d_in[6];
    const float* bk         = (const float*)d_in[7];
    const float* Wc         = (const float*)d_in[8];
    const float* bc         = (const float*)d_in[9];
    const float* Wp         = (const float*)d_in[10];
    const float* bp         = (const float*)d_in[11];
    const float* W_ih0      = (const float*)d_in[12];
    const float* W_hh0      = (const float*)d_in[13];
    const float* b_ih0      = (const float*)d_in[14];
    const float* b_hh0      = (const float*)d_in[15];
    const float* W_ih1      = (const float*)d_in[16];
    const float* W_hh1      = (const float*)d_in[17];
    const float* b_ih1      = (const float*)d_in[18];
    const float* b_hh1      = (const float*)d_in[19];
    float* out = (float*)d_out;

    // Workspace partitioning (floats)
    float* ws    = (float*)d_ws;
    float* attk  = ws;                         // S*B*H
    float* hs    = attk + (long)SSZ*BSZ*HSZ;   // L*B*H
    float* cs    = hs   + (long)LSZ*BSZ*HSZ;   // L*B*H
    float* feed  = cs   + (long)LSZ*BSZ*HSZ;   // B*H   (== comb)
    float* xbuf  = feed + (long)BSZ*HSZ;       // B*(E+H)
    float* gbuf  = xbuf + (long)BSZ*(ESZ+HSZ); // B*4H
    float* cat   = gbuf + (long)BSZ*4*HSZ;     // B*2H
    float* dist  = cat  + (long)BSZ*2*HSZ;     // S*B
    float* plog  = dist + (long)SSZ*BSZ;       // B*V
    float* copyp = plog + (long)BSZ*VSZ;       // B*V

    float* h_l0 = hs, *h_l1 = hs + BSZ*HSZ;
    float* c_l0 = cs, *c_l1 = cs + BSZ*HSZ;

    // Init state + feed=0
    init_state_kernel<<<(LSZ*BSZ*HSZ + 255)/256, 256, 0, stream>>>(h0, c0, hs, cs, feed);

    // att_key = enc @ Wk^T + bk   [S*B, H] x [H, H]   (32x32 tiles)
    gemm_bias_kernel<<<dim3(HSZ/32, (SSZ*BSZ)/32), 32, 0, stream>>>(
        enc, HSZ, Wk, HSZ, bk, attk, HSZ, HSZ, HSZ);

    for (int t = 0; t < TSZ; ++t) {
        embed_concat_kernel<<<BSZ, 256, 0, stream>>>(ref_tokens, t, embed, feed, xbuf);

        // Layer 0 gates + update
        gates_kernel<<<(4*HSZ)/32, 32, 0, stream>>>(
            xbuf, ESZ+HSZ, W_ih0, ESZ+HSZ, h_l0, W_hh0, b_ih0, b_hh0, gbuf);
        lstm_update_kernel<<<(BSZ*HSZ)/256, 256, 0, stream>>>(gbuf, h_l0, c_l0);

        // Layer 1 gates + update (input = new h_l0, recurrent = old h_l1)
        gates_kernel<<<(4*HSZ)/32, 32, 0, stream>>>(
            h_l0, HSZ, W_ih1, HSZ, h_l1, W_hh1, b_ih1, b_hh1, gbuf);
        lstm_update_kernel<<<(BSZ*HSZ)/256, 256, 0, stream>>>(gbuf, h_l1, c_l1);

        // Attention
        attn_scores_kernel<<<(SSZ*BSZ)/8, 256, 0, stream>>>(attk, h_l1, src_tokens, dist);
        attn_combine_kernel<<<BSZ, 256, 0, stream>>>(dist, enc, h_l1, cat);

        // comb = cat @ Wc^T + bc  -> also next step's feed
        gemm_bias_kernel<<<dim3(HSZ/32, 1), 32, 0, stream>>>(
            cat, 2*HSZ, Wc, 2*HSZ, bc, feed, HSZ, 2*HSZ, HSZ);

        // pred_logits = comb @ Wp^T + bp   (N = 10000 = 625*16 -> 313 blocks)
        gemm_bias_kernel<<<dim3((VSZ + 31)/32, 1), 32, 0, stream>>>(
            feed, HSZ, Wp, HSZ, bp, plog, VSZ, HSZ, VSZ);

        // copy mechanism + output
        copy_probs_kernel<<<BSZ, 256, 0, stream>>>(src_tokens, dist, copyp);
        final_out_kernel<<<BSZ, 256, 0, stream>>>(plog, copyp, out + (long)t*BSZ*VSZ);
    }
}